// MaskedNorm_33320356282917
// MI455X (gfx1250) — compile-verified
//
#include <hip/hip_runtime.h>
#include <hip/hip_bf16.h>

// ---------------------------------------------------------------------------
// MaskedNorm for MI455X (gfx1250).
// Memory-bound: min traffic ~270MB (y read for stats from HBM, y re-read for
// apply from L2 [134MB < 192MB L2], out written NT) => ~12us at 23.3 TB/s.
// Pass 1: async global->LDS staging + WMMA f32 16x16x4 masked column sums.
// Pass 2: tiny finalize (n, scale, mean).
// Pass 3: float4 streaming apply with non-temporal output stores.
// ---------------------------------------------------------------------------

typedef __attribute__((ext_vector_type(2))) float v2f;
typedef __attribute__((ext_vector_type(4))) float v4f;
typedef __attribute__((ext_vector_type(8))) float v8f;

#define B_DIM 8
#define T_DIM 4096
#define C_DIM 1024
#define NROWS (B_DIM * T_DIM)            // 32768 tokens
#define TILE_ROWS 16
#define SLICE_CH 256                     // channels per block (grid.y = 4)
#define ROWS_PER_BLOCK 512               // 32 tiles of 16 rows
#define NTILES (ROWS_PER_BLOCK / TILE_ROWS)

// ---------------------------------------------------------------------------
// Kernel 0: zero the S1/S2 accumulators in workspace.
// ---------------------------------------------------------------------------
__global__ __launch_bounds__(256) void mn_zero_kernel(float* __restrict__ ws) {
    int i = blockIdx.x * 256 + threadIdx.x;   // 2048 floats
    if (i < 2 * C_DIM) ws[i] = 0.0f;
}

// ---------------------------------------------------------------------------
// Kernel 1: masked per-channel sum + sum-of-squares via WMMA, fed by
// double-buffered async global->LDS copies.
//   grid = (NROWS/ROWS_PER_BLOCK, C_DIM/SLICE_CH), block = 256 (8 waves).
//   Wave w owns 32 channels (two 16-ch WMMA tiles) of the 256-ch slice.
//   A operand carries the row mask => masking happens inside the matrix unit.
// ---------------------------------------------------------------------------
__global__ __launch_bounds__(256) void mn_stats_kernel(
    const float* __restrict__ y, const int* __restrict__ mask,
    float* __restrict__ S1, float* __restrict__ S2)
{
    __shared__ float ybuf[2][TILE_ROWS * SLICE_CH];   // 2 x 16KB
    __shared__ float smask[ROWS_PER_BLOCK];           // 2KB

    const int tid    = threadIdx.x;
    const int waveId = tid >> 5;
    const int lane   = tid & 31;
    const int row0   = blockIdx.x * ROWS_PER_BLOCK;
    const int cbase  = blockIdx.y * SLICE_CH;

    // Stage this strip's mask (0/1 floats) into LDS.
    for (int i = tid; i < ROWS_PER_BLOCK; i += 256)
        smask[i] = (mask[row0 + i] > 0) ? 1.0f : 0.0f;

    // LDS byte addresses (low 32 bits of the generic pointer = workgroup-
    // relative LDS offset, which is what GLOBAL_LOAD_ASYNC_TO_LDS expects).
    const unsigned int lb0 = (unsigned int)(unsigned long long)(&ybuf[0][0]);
    const unsigned int lb1 = (unsigned int)(unsigned long long)(&ybuf[1][0]);

    // Issue one 16-row x 256-ch tile: 1024 b128 chunks, 4 per thread.
    auto issue_tile = [&](int k) {
        const unsigned int lbase = (k & 1) ? lb1 : lb0;
        const float* gbase = y + (size_t)(row0 + k * TILE_ROWS) * C_DIM + cbase;
#pragma unroll
        for (int it = 0; it < 4; ++it) {
            int chunk = it * 256 + tid;       // 0..1023 float4 chunks
            int r  = chunk >> 6;              // 64 float4 per 256-ch row
            int cq = chunk & 63;
            const float* gp = gbase + (size_t)r * C_DIM + cq * 4;
            unsigned int laddr = lbase + (unsigned int)chunk * 16u;
            asm volatile("global_load_async_to_lds_b128 %0, %1, off"
                         :: "v"(laddr), "v"((unsigned long long)gp)
                         : "memory");
        }
    };

    v8f acc1[2] = {};   // S1 accumulators (two 16-ch tiles)
    v8f acc2[2] = {};   // S2 accumulators

    issue_tile(0);
    __syncthreads();    // smask visible to all waves

    const int kb = (lane >> 4) << 1;          // lane<16 -> K=0,1 ; else K=2,3
    const int cl = lane & 15;

    for (int k = 0; k < NTILES; ++k) {
        if (k + 1 < NTILES) {
            issue_tile(k + 1);
            // 4 outstanding = next tile only => tile k fully landed (per wave).
            asm volatile("s_wait_asynccnt 0x4" ::: "memory");
        } else {
            asm volatile("s_wait_asynccnt 0x0" ::: "memory");
        }
        __syncthreads();                      // tile k visible block-wide

        const float* buf = (k & 1) ? &ybuf[1][0] : &ybuf[0][0];
        const int mrow = k * TILE_ROWS;
#pragma unroll
        for (int s = 0; s < 4; ++s) {
            // A[i][k'] = mask(row 4s+k') for all i: lanes<16 carry K=0,1.
            v2f a;
            a.x = smask[mrow + 4 * s + kb + 0];
            a.y = smask[mrow + 4 * s + kb + 1];
#pragma unroll
            for (int ct = 0; ct < 2; ++ct) {
                int c = waveId * 32 + ct * 16 + cl;
                float y0 = buf[(4 * s + kb + 0) * SLICE_CH + c];
                float y1 = buf[(4 * s + kb + 1) * SLICE_CH + c];
                v2f b;  b.x  = y0;        b.y  = y1;
                v2f b2; b2.x = y0 * y0;   b2.y = y1 * y1;
                acc1[ct] = __builtin_amdgcn_wmma_f32_16x16x4_f32(
                    false, a, false, b,  (short)0, acc1[ct], false, false);
                acc2[ct] = __builtin_amdgcn_wmma_f32_16x16x4_f32(
                    false, a, false, b2, (short)0, acc2[ct], false, false);
            }
        }
        __syncthreads();                      // buffer k reusable at k+2
    }

    // D rows are identical (A rows identical); lane n, VGPR0 holds channel n.
    if (lane < 16) {
#pragma unroll
        for (int ct = 0; ct < 2; ++ct) {
            int c = cbase + waveId * 32 + ct * 16 + lane;
            atomicAdd(&S1[c], acc1[ct][0]);
            atomicAdd(&S2[c], acc2[ct][0]);
        }
    }
}

// ---------------------------------------------------------------------------
// Kernel 2: single block. Reduce n, then per-channel scale/mean.
//   scale[c] = gamma[c] / (std[c] + eps);  mean[c] stored for apply.
// ---------------------------------------------------------------------------
__global__ __launch_bounds__(1024) void mn_finalize_kernel(
    const int* __restrict__ mask, const float* __restrict__ gamma,
    float* __restrict__ ws)
{
    __shared__ float red[1024];
    const int tid = threadIdx.x;
    float cnt = 0.0f;
    for (int i = tid; i < NROWS; i += 1024)
        cnt += (mask[i] > 0) ? 1.0f : 0.0f;
    red[tid] = cnt;
    __syncthreads();
    for (int s = 512; s > 0; s >>= 1) {
        if (tid < s) red[tid] += red[tid + s];
        __syncthreads();
    }
    const float n = red[0];

    const float s1 = ws[tid];
    const float s2 = ws[C_DIM + tid];
    const float mean = s1 / n;
    float var = (s2 - n * mean * mean) / (n - 1.0f);
    var = fmaxf(var, 0.0f);
    const float stddev = sqrtf(var);
    const float sc = gamma[tid] / (stddev + 1e-4f);
    ws[2 * C_DIM + tid] = sc;
    ws[3 * C_DIM + tid] = mean;
}

// ---------------------------------------------------------------------------
// Kernel 3: streaming apply. One block == one token row (256 float4 = 1024 ch),
// so the mask load is wave-uniform (scalar). y re-read likely hits L2; output
// stored non-temporally so it does not evict y from L2 mid-pass.
//   out = m ? (y - mean) * scale + beta : y
// ---------------------------------------------------------------------------
__global__ __launch_bounds__(256) void mn_apply_kernel(
    const float* __restrict__ y, const int* __restrict__ mask,
    const float* __restrict__ ws, const float* __restrict__ beta,
    float* __restrict__ out)
{
    const int row = blockIdx.x;                 // token index 0..32767
    const int c4  = threadIdx.x;                // float4 index within row
    const int idx = row * 256 + c4;

    v4f v = reinterpret_cast<const v4f*>(y)[idx];
    if (mask[row] > 0) {                        // uniform -> scalar branch
        const v4f sc = reinterpret_cast<const v4f*>(ws + 2 * C_DIM)[c4];
        const v4f mn = reinterpret_cast<const v4f*>(ws + 3 * C_DIM)[c4];
        const v4f bt = reinterpret_cast<const v4f*>(beta)[c4];
        v.x = fmaf(v.x - mn.x, sc.x, bt.x);
        v.y = fmaf(v.y - mn.y, sc.y, bt.y);
        v.z = fmaf(v.z - mn.z, sc.z, bt.z);
        v.w = fmaf(v.w - mn.w, sc.w, bt.w);
    }
    __builtin_nontemporal_store(v, reinterpret_cast<v4f*>(out) + idx);
}

// ---------------------------------------------------------------------------
extern "C" void kernel_launch(void* const* d_in, const int* in_sizes, int n_in,
                              void* d_out, int out_size, void* d_ws, size_t ws_size,
                              hipStream_t stream) {
    (void)in_sizes; (void)n_in; (void)out_size; (void)ws_size;
    const float* y     = (const float*)d_in[0];
    const int*   mask  = (const int*)d_in[1];
    const float* gamma = (const float*)d_in[2];
    const float* beta  = (const float*)d_in[3];
    float* out = (float*)d_out;
    float* ws  = (float*)d_ws;   // [S1 | S2 | scale | mean], 4*1024 floats

    mn_zero_kernel<<<8, 256, 0, stream>>>(ws);

    dim3 sgrid(NROWS / ROWS_PER_BLOCK, C_DIM / SLICE_CH);   // (64, 4)
    mn_stats_kernel<<<sgrid, 256, 0, stream>>>(y, mask, ws, ws + C_DIM);

    mn_finalize_kernel<<<1, 1024, 0, stream>>>(mask, gamma, ws);

    mn_apply_kernel<<<NROWS, 256, 0, stream>>>(y, mask, ws, beta, out);
}